// DisenGCNLayer_31937376813402
// MI455X (gfx1250) — compile-verified
//
#include <hip/hip_runtime.h>
#include <hip/hip_bf16.h>

typedef __attribute__((ext_vector_type(2))) float v2f;
typedef __attribute__((ext_vector_type(8))) float v8f;

#define IN_FEATS 128
#define OUT_FEATS 64
#define KFAC 8
#define FDIM 8
#define LDS_STRIDE 130   // padded stride (floats): conflict-free ds_load_b64

// ---------------------------------------------------------------------------
// GEMM: h_raw = leaky_relu(x @ W + bias), using V_WMMA_F32_16X16X4_F32.
// Block = 128 threads = 4 waves; block handles a 16-row stripe, wave w owns
// the 16x16 tile covering columns [16w, 16w+16).
// ---------------------------------------------------------------------------
__global__ __launch_bounds__(128) void gemm_bias_lrelu_kernel(
    const float* __restrict__ x, const float* __restrict__ W,
    const float* __restrict__ bias, float* __restrict__ h_raw) {
  __shared__ float Alds[16 * LDS_STRIDE];          // 16 x 128 (padded)
  __shared__ float Wlds[OUT_FEATS * LDS_STRIDE];   // transposed: [col][k]

  const int t = threadIdx.x;       // 0..127
  const int wave = t >> 5;         // 0..3
  const int lane = t & 31;
  const int row0 = blockIdx.x * 16;

  // Stage A tile (16 x 128 floats = 512 float4), coalesced global reads.
  {
    const float4* xg = (const float4*)(x + (size_t)row0 * IN_FEATS);
#pragma unroll
    for (int i = 0; i < 4; ++i) {
      int idx = t + 128 * i;             // float4 index 0..511
      int r = idx >> 5;                  // row 0..15
      int c4 = idx & 31;                 // float4 within row
      float4 v = xg[r * 32 + c4];
      float* dp = &Alds[r * LDS_STRIDE + c4 * 4];
      dp[0] = v.x; dp[1] = v.y; dp[2] = v.z; dp[3] = v.w;
    }
  }
  // Stage W transposed: Wlds[c * LDS_STRIDE + k] = W[k*64 + c].
  {
    const float4* wg = (const float4*)W;  // 128 rows x 16 float4
#pragma unroll
    for (int i = 0; i < 16; ++i) {
      int idx = t + 128 * i;             // 0..2047
      int kk = idx >> 4;                 // k row 0..127
      int c4 = idx & 15;
      float4 v = wg[idx];
      int c = c4 * 4;
      Wlds[(c + 0) * LDS_STRIDE + kk] = v.x;
      Wlds[(c + 1) * LDS_STRIDE + kk] = v.y;
      Wlds[(c + 2) * LDS_STRIDE + kk] = v.z;
      Wlds[(c + 3) * LDS_STRIDE + kk] = v.w;
    }
  }
  __syncthreads();

  // 16x16x4 f32 A layout: lanes 0-15 hold M=lane, K={k0,k0+1};
  // lanes 16-31 hold M=lane-16, K={k0+2,k0+3}. B mirrors with N.
  const int half = lane >> 4;            // 0/1
  const int m = lane & 15;
  const int col = (wave << 4) + m;       // 0..63
  const float* ap = &Alds[m * LDS_STRIDE + 2 * half];
  const float* bp = &Wlds[col * LDS_STRIDE + 2 * half];

  v8f acc = {0.f, 0.f, 0.f, 0.f, 0.f, 0.f, 0.f, 0.f};
#pragma unroll
  for (int k0 = 0; k0 < IN_FEATS; k0 += 4) {
    v2f a; a.x = ap[k0]; a.y = ap[k0 + 1];
    v2f b; b.x = bp[k0]; b.y = bp[k0 + 1];
    acc = __builtin_amdgcn_wmma_f32_16x16x4_f32(
        false, a, false, b, (short)0, acc, false, false);
  }

  // C layout: VGPR v -> M = v + 8*half, N = lane&15 (within tile).
  const float bv = bias[col];
#pragma unroll
  for (int v = 0; v < 8; ++v) {
    int r = row0 + v + 8 * half;
    float val = acc[v] + bv;
    val = (val >= 0.f) ? val : val * 0.01f;   // leaky ReLU
    h_raw[(size_t)r * OUT_FEATS + col] = val;
  }
}

// ---------------------------------------------------------------------------
// Per-factor L2 normalize: thread per (node, factor) over 8 floats.
// ---------------------------------------------------------------------------
__global__ void l2norm_kernel(const float* __restrict__ hraw,
                              float* __restrict__ hn, int NK) {
  int tid = blockIdx.x * blockDim.x + threadIdx.x;
  if (tid >= NK) return;
  const float4* p = (const float4*)(hraw + (size_t)tid * FDIM);
  float4 a = p[0], b = p[1];
  float ss = a.x * a.x + a.y * a.y + a.z * a.z + a.w * a.w +
             b.x * b.x + b.y * b.y + b.z * b.z + b.w * b.w;
  float inv = rsqrtf(ss);
  a.x *= inv; a.y *= inv; a.z *= inv; a.w *= inv;
  b.x *= inv; b.y *= inv; b.z *= inv; b.w *= inv;
  float4* o = (float4*)(hn + (size_t)tid * FDIM);
  o[0] = a; o[1] = b;
}

// ---------------------------------------------------------------------------
// Edge pass: thread per (edge, factor). s = <h_dst[src], h_normed[dst]>/tau,
// ex = exp(s)  (|s|<=1 since factors are unit vectors -> no max-stabilization
// needed). Accumulate denom[src,k] += ex, num[src,k,:] += ex * feat_nbr.
// ---------------------------------------------------------------------------
__global__ void edge_kernel(const int* __restrict__ ei,
                            const float* __restrict__ hn,   // h_src (fixed)
                            const float* __restrict__ hd,   // h_dst (current)
                            float* __restrict__ num, float* __restrict__ den,
                            int E) {
  int tid = blockIdx.x * blockDim.x + threadIdx.x;
  if (tid >= E * KFAC) return;
  int e = tid >> 3;
  int k = tid & 7;
  int sn = ei[e];        // src (segment node)
  int dn = ei[E + e];    // dst (gathered neighbor)

  const float4* pa = (const float4*)(hd + (size_t)sn * OUT_FEATS + k * FDIM);
  const float4* pb = (const float4*)(hn + (size_t)dn * OUT_FEATS + k * FDIM);
  float4 a0 = pa[0], a1 = pa[1];
  float4 b0 = pb[0], b1 = pb[1];

  float s = a0.x * b0.x + a0.y * b0.y + a0.z * b0.z + a0.w * b0.w +
            a1.x * b1.x + a1.y * b1.y + a1.z * b1.z + a1.w * b1.w;
  float ex = __expf(s);  // TAU = 1

  atomicAdd(&den[(size_t)sn * KFAC + k], ex);
  float* np = num + (size_t)sn * OUT_FEATS + k * FDIM;
  atomicAdd(np + 0, ex * b0.x);
  atomicAdd(np + 1, ex * b0.y);
  atomicAdd(np + 2, ex * b0.z);
  atomicAdd(np + 3, ex * b0.w);
  atomicAdd(np + 4, ex * b1.x);
  atomicAdd(np + 5, ex * b1.y);
  atomicAdd(np + 6, ex * b1.z);
  atomicAdd(np + 7, ex * b1.w);
}

// ---------------------------------------------------------------------------
// Finalize: node_attr = num/denom + h_normed, re-normalize, write h_dst (or
// final output). Also re-zeros num/denom for the next iteration.
// ---------------------------------------------------------------------------
__global__ void finalize_kernel(float* __restrict__ num, float* __restrict__ den,
                                const float* __restrict__ hn,
                                float* __restrict__ out, int NK) {
  int tid = blockIdx.x * blockDim.x + threadIdx.x;
  if (tid >= NK) return;
  float d = den[tid];
  float invd = (d != 0.f) ? 1.f / d : 0.f;   // no-incoming-edge guard
  float4* np = (float4*)(num + (size_t)tid * FDIM);
  const float4* hp = (const float4*)(hn + (size_t)tid * FDIM);
  float4 n0 = np[0], n1 = np[1];
  float4 h0 = hp[0], h1 = hp[1];
  float v0 = n0.x * invd + h0.x, v1 = n0.y * invd + h0.y;
  float v2 = n0.z * invd + h0.z, v3 = n0.w * invd + h0.w;
  float v4 = n1.x * invd + h1.x, v5 = n1.y * invd + h1.y;
  float v6 = n1.z * invd + h1.z, v7 = n1.w * invd + h1.w;
  float ss = v0 * v0 + v1 * v1 + v2 * v2 + v3 * v3 +
             v4 * v4 + v5 * v5 + v6 * v6 + v7 * v7;
  float inv = rsqrtf(ss);
  float4 o0 = {v0 * inv, v1 * inv, v2 * inv, v3 * inv};
  float4 o1 = {v4 * inv, v5 * inv, v6 * inv, v7 * inv};
  float4* op = (float4*)(out + (size_t)tid * FDIM);
  op[0] = o0; op[1] = o1;
  // zero accumulators for next iteration
  float4 z = {0.f, 0.f, 0.f, 0.f};
  np[0] = z; np[1] = z;
  den[tid] = 0.f;
}

extern "C" void kernel_launch(void* const* d_in, const int* in_sizes, int n_in,
                              void* d_out, int out_size, void* d_ws, size_t ws_size,
                              hipStream_t stream) {
  const float* x = (const float*)d_in[0];
  const float* W = (const float*)d_in[1];
  const float* bias = (const float*)d_in[2];
  const int* ei = (const int*)d_in[3];
  float* out = (float*)d_out;

  const int N = in_sizes[0] / IN_FEATS;   // 40000
  const int E = in_sizes[3] / 2;          // 640000
  const int NK = N * KFAC;

  float* ws = (float*)d_ws;
  float* h_normed = ws;                        // N*64
  float* h_dst    = ws + (size_t)N * 64;       // N*64 (also staging for h_raw)
  float* num      = ws + (size_t)2 * N * 64;   // N*64
  float* den      = ws + (size_t)3 * N * 64;   // N*8  (contiguous after num)

  // zero num + denom (contiguous region)
  hipMemsetAsync(num, 0, (size_t)(N * 64 + N * 8) * sizeof(float), stream);

  // GEMM + bias + leaky ReLU -> h_raw (staged in h_dst buffer)
  gemm_bias_lrelu_kernel<<<N / 16, 128, 0, stream>>>(x, W, bias, h_dst);

  // per-factor L2 normalize -> h_normed
  l2norm_kernel<<<(NK + 255) / 256, 256, 0, stream>>>(h_dst, h_normed, NK);

  for (int it = 0; it < 3; ++it) {
    const float* hd = (it == 0) ? h_normed : h_dst;   // h_dst starts at h_normed
    edge_kernel<<<(E * KFAC + 255) / 256, 256, 0, stream>>>(
        ei, h_normed, hd, num, den, E);
    float* outp = (it == 2) ? out : h_dst;
    finalize_kernel<<<(NK + 255) / 256, 256, 0, stream>>>(
        num, den, h_normed, outp, NK);
  }
}